// GCN_45097156608207
// MI455X (gfx1250) — compile-verified
//
#include <hip/hip_runtime.h>

// GCN (2-layer) for MI455X / gfx1250.
// GEMMs use V_WMMA_F32_16X16X4_F32 (one 16x16 C tile per wave32).
// Edge aggregation = streaming gather + global_atomic_add_f32 (L2-resident).

typedef __attribute__((ext_vector_type(2))) float v2f;
typedef __attribute__((ext_vector_type(8))) float v8f;

#define HIDDEN   16
#define EMBED    128
#define NFEAT    128
#define NGRAPHS  64

// ---------------- utility fills ----------------
__global__ void k_fill(float* __restrict__ p, long long n, float v) {
  long long i = (long long)blockIdx.x * blockDim.x + threadIdx.x;
  if (i < n) p[i] = v;
}

// ---------------- degree / dinv ----------------
__global__ void k_degree(const int* __restrict__ dst, float* __restrict__ deg, int nE) {
  int e = blockIdx.x * blockDim.x + threadIdx.x;
  if (e < nE) atomicAdd(&deg[dst[e]], 1.0f);
}

__global__ void k_dinv(float* __restrict__ dinv, int n) {
  int i = blockIdx.x * blockDim.x + threadIdx.x;
  if (i < n) {
    float d = dinv[i];
    dinv[i] = (d > 0.0f) ? rsqrtf(d) : 0.0f;
  }
}

// ---------------- WMMA f32 GEMM: C[MxN] = A[MxK] * B[KxN] ----------------
// M multiple of 16, N multiple of 16, K multiple of 4. One wave per C tile.
__global__ void k_gemm_wmma(const float* __restrict__ A, const float* __restrict__ B,
                            float* __restrict__ C, int nRowTiles, int nColTiles,
                            int K, int N) {
  int wave = (int)((blockIdx.x * blockDim.x + threadIdx.x) >> 5);
  if (wave >= nRowTiles * nColTiles) return;   // wave-uniform: EXEC stays all-1s
  int lane = threadIdx.x & 31;
  int rowTile = wave / nColTiles;
  int colTile = wave - rowTile * nColTiles;
  int row0 = rowTile << 4;
  int col0 = colTile << 4;
  int l  = lane & 15;   // M index for A, N index for B/C
  int hi = lane >> 4;   // selects K pair (A/B) and M half (C/D)

  v8f acc = {};
  const float* arow = A + (size_t)(row0 + l) * K;
  for (int k0 = 0; k0 < K; k0 += 4) {
    int kb = k0 + 2 * hi;
    v2f a, b;
    a.x = arow[kb];
    a.y = arow[kb + 1];
    b.x = B[(size_t)kb * N + col0 + l];
    b.y = B[(size_t)(kb + 1) * N + col0 + l];
    // (neg_a, A, neg_b, B, c_mod, C, reuse_a, reuse_b)
    acc = __builtin_amdgcn_wmma_f32_16x16x4_f32(false, a, false, b,
                                                (short)0, acc, false, false);
  }
  // D layout: VGPR r -> row (row0 + r + 8*hi), col (col0 + l)
  #pragma unroll
  for (int r = 0; r < 8; ++r) {
    C[(size_t)(row0 + r + 8 * hi) * N + (col0 + l)] = acc[r];
  }
}

// ---------------- edge aggregation: agg[dst] += h[src] * dinv[s]*dinv[d] ----------------
// one thread per (edge, feature); F = 1<<logF consecutive lanes share an edge.
__global__ void k_aggregate(const float* __restrict__ h, const int* __restrict__ src,
                            const int* __restrict__ dst, const float* __restrict__ dinv,
                            float* __restrict__ agg, int nE, int logF) {
  long long tid = (long long)blockIdx.x * blockDim.x + threadIdx.x;
  long long total = (long long)nE << logF;
  if (tid >= total) return;
  int e = (int)(tid >> logF);
  int f = (int)(tid & ((1 << logF) - 1));
  int s = src[e];
  int d = dst[e];
  float w = dinv[s] * dinv[d];
  atomicAdd(&agg[((size_t)d << logF) + f], h[((size_t)s << logF) + f] * w);
}

// ---------------- layer-1 finalize: add self-loop + bias, ReLU (in place into agg) ----
__global__ void k_finalize1(const float* __restrict__ hpre, float* __restrict__ agg,
                            const float* __restrict__ dinv, const float* __restrict__ bias,
                            int nNodes) {
  int tid = blockIdx.x * blockDim.x + threadIdx.x;
  if (tid >= nNodes * HIDDEN) return;
  int i = tid >> 4;
  int f = tid & (HIDDEN - 1);
  float di = dinv[i];
  float v = agg[tid] + hpre[tid] * di * di + bias[f];
  agg[tid] = fmaxf(v, 0.0f);
}

// ---------------- layer-2 finalize fused with mean-pool accumulation ----------------
__global__ void k_finalize2(const float* __restrict__ hpre, const float* __restrict__ agg,
                            const float* __restrict__ dinv, const float* __restrict__ bias,
                            const int* __restrict__ batch, float* __restrict__ pool,
                            int nNodes) {
  long long tid = (long long)blockIdx.x * blockDim.x + threadIdx.x;
  if (tid >= (long long)nNodes * EMBED) return;
  int i = (int)(tid >> 7);
  int f = (int)(tid & (EMBED - 1));
  float di = dinv[i];
  float v = agg[tid] + hpre[tid] * di * di + bias[f];
  atomicAdd(&pool[(size_t)batch[i] * EMBED + f], v);
}

__global__ void k_count(const int* __restrict__ batch, float* __restrict__ cnt, int nNodes) {
  int i = blockIdx.x * blockDim.x + threadIdx.x;
  if (i < nNodes) atomicAdd(&cnt[batch[i]], 1.0f);
}

__global__ void k_out(const float* __restrict__ pool, const float* __restrict__ cnt,
                      float* __restrict__ out) {
  int tid = blockIdx.x * blockDim.x + threadIdx.x;
  if (tid >= NGRAPHS * EMBED) return;
  int g = tid >> 7;
  out[tid] = pool[tid] / fmaxf(cnt[g], 1.0f);
}

// ---------------- host launcher ----------------
static inline unsigned nblk(long long n, int t) { return (unsigned)((n + t - 1) / t); }

extern "C" void kernel_launch(void* const* d_in, const int* in_sizes, int n_in,
                              void* d_out, int out_size, void* d_ws, size_t ws_size,
                              hipStream_t stream) {
  const float* x     = (const float*)d_in[0];
  const int*   ei    = (const int*)d_in[1];
  const int*   batch = (const int*)d_in[2];
  const float* W1    = (const float*)d_in[3];
  const float* b1    = (const float*)d_in[4];
  const float* W2    = (const float*)d_in[5];
  const float* b2    = (const float*)d_in[6];
  float* out = (float*)d_out;

  const int nNodes = in_sizes[2];        // 100000 (divisible by 16)
  const int nE     = in_sizes[1] / 2;    // 1,600,000
  const int* src = ei;
  const int* dst = ei + nE;

  float* ws = (float*)d_ws;
  size_t off = 0;
  float* dinv = ws + off; off += (size_t)nNodes;
  float* h1p  = ws + off; off += (size_t)nNodes * HIDDEN;
  float* agg1 = ws + off; off += (size_t)nNodes * HIDDEN;   // becomes h1 after ReLU
  float* h2p  = ws + off; off += (size_t)nNodes * EMBED;
  float* agg2 = ws + off; off += (size_t)nNodes * EMBED;
  float* pool = ws + off; off += (size_t)NGRAPHS * EMBED;
  float* cnt  = ws + off; off += (size_t)NGRAPHS;           // pool & cnt contiguous

  const int T = 256;

  // init: deg starts at 1.0 (self-loop); accumulators at 0
  k_fill<<<nblk(nNodes, T), T, 0, stream>>>(dinv, nNodes, 1.0f);
  k_fill<<<nblk((long long)nNodes * HIDDEN, T), T, 0, stream>>>(agg1, (long long)nNodes * HIDDEN, 0.0f);
  k_fill<<<nblk((long long)nNodes * EMBED, T), T, 0, stream>>>(agg2, (long long)nNodes * EMBED, 0.0f);
  k_fill<<<nblk(NGRAPHS * EMBED + NGRAPHS, T), T, 0, stream>>>(pool, NGRAPHS * EMBED + NGRAPHS, 0.0f);

  // degree / D^{-1/2}
  k_degree<<<nblk(nE, T), T, 0, stream>>>(dst, dinv, nE);
  k_dinv<<<nblk(nNodes, T), T, 0, stream>>>(dinv, nNodes);

  const int rowTiles = nNodes / 16;

  // layer 1: h1p = x @ W1  (100000x128 @ 128x16)
  {
    long long waves = (long long)rowTiles * (HIDDEN / 16);
    k_gemm_wmma<<<nblk(waves * 32, T), T, 0, stream>>>(x, W1, h1p, rowTiles, HIDDEN / 16, NFEAT, HIDDEN);
  }
  k_aggregate<<<nblk((long long)nE * HIDDEN, T), T, 0, stream>>>(h1p, src, dst, dinv, agg1, nE, 4);
  k_finalize1<<<nblk((long long)nNodes * HIDDEN, T), T, 0, stream>>>(h1p, agg1, dinv, b1, nNodes);

  // layer 2: h2p = relu(h1) @ W2  (100000x16 @ 16x128)
  {
    long long waves = (long long)rowTiles * (EMBED / 16);
    k_gemm_wmma<<<nblk(waves * 32, T), T, 0, stream>>>(agg1, W2, h2p, rowTiles, EMBED / 16, HIDDEN, EMBED);
  }
  k_aggregate<<<nblk((long long)nE * EMBED, T), T, 0, stream>>>(h2p, src, dst, dinv, agg2, nE, 7);

  // mean pooling (fused with bias + self-loop finalize)
  k_count<<<nblk(nNodes, T), T, 0, stream>>>(batch, cnt, nNodes);
  k_finalize2<<<nblk((long long)nNodes * EMBED, T), T, 0, stream>>>(h2p, agg2, dinv, b2, batch, pool, nNodes);
  k_out<<<nblk(NGRAPHS * EMBED, T), T, 0, stream>>>(pool, cnt, out);
}